// MoChAAttention_67851893342406
// MI455X (gfx1250) — compile-verified
//
#include <hip/hip_runtime.h>
#include <hip/hip_bf16.h>

// MoChA stable chunkwise attention, CHUNK = 16, on gfx1250 (MI455X).
//
// out[b,t] = exp(logits[b,t]) * sum_{s=t..t+15} r[b,s]
//   r[b,s] = emit[b,s] * exp(-max[b,s]) / denom[b,s]
// The 16-wide sliding sum is computed as a banded matmul with a constant
// 0/1 16x32 band matrix via eight chained V_WMMA_F32_16X16X4_F32.

typedef __attribute__((ext_vector_type(2))) float v2f;
typedef __attribute__((ext_vector_type(8))) float v8f;

static constexpr int   kT    = 16384;   // time length (fixed by reference)
static constexpr int   kW    = 16;      // CHUNK window
static constexpr int   kSeg  = 2048;    // columns per workgroup
static constexpr int   kThr  = 256;     // 8 wave32 per block
static constexpr float kFMax = 3.402823466e+38f;

__global__ __launch_bounds__(kThr) void mocha_chunk_attn_kernel(
    const float* __restrict__ emit,     // [B, T]
    const float* __restrict__ logits,   // [B, T]
    float* __restrict__ out)            // [B, T]
{
    const int segs_per_row = kT / kSeg;                 // 8
    const int b   = blockIdx.x / segs_per_row;
    const int c0  = (blockIdx.x % segs_per_row) * kSeg; // first t of this block

    const float* row_logits = logits + (size_t)b * kT;
    const float* row_emit   = emit   + (size_t)b * kT;
    float*       row_out    = out    + (size_t)b * kT;

    // s_logits[q] = logits[c0 - 15 + q], q in [0, kSeg+31)
    __shared__ float s_logits[kSeg + 2 * kW];           // 2080 floats
    __shared__ float s_r[kSeg + kW];                    // 2064 floats

    // ---- Phase 1a: stage logits (+15 halo each side) into LDS ----
    for (int q = threadIdx.x; q < kSeg + 2 * kW - 1; q += kThr) {
        const int c = c0 - (kW - 1) + q;
        s_logits[q] = (c >= 0 && c < kT) ? row_logits[c] : -kFMax;
    }
    __syncthreads();

    // ---- Phase 1b: per-column window max, denom, and r[s] ----
    for (int ls = threadIdx.x; ls < kSeg + kW; ls += kThr) {
        const int c = c0 + ls;                          // global column of s
        float r = 0.0f;
        if (c < kT) {
            // window = logits[c-15..c] = s_logits[ls .. ls+15]
            float m = -kFMax;
#pragma unroll
            for (int k = 0; k < kW; ++k) m = fmaxf(m, s_logits[ls + k]);
            float den = 0.0f;
#pragma unroll
            for (int k = 0; k < kW; ++k) den += __expf(s_logits[ls + k] - m);
            r = row_emit[c] * __expf(-m) / den;
        }
        s_r[ls] = r;                                    // halo past row end = 0
    }
    __syncthreads();

    // ---- Phase 2: banded sliding-window sum via WMMA f32 16x16x4 ----
    // Wave wv produces 256 outputs t_local in [256*wv, 256*wv+256).
    // D[i,n] = sum_j A[i,j] * B[j,n],  A[i,j] = (0 <= j-i <= 15),
    // B[j,n] = r[wbase + 16*n + j],  j in [0,32) split into 8 K=4 slices.
    const int lane  = threadIdx.x & 31;
    const int wv    = threadIdx.x >> 5;                 // 0..7
    const int wbase = wv * 256;                         // base into s_r
    const int col   = lane & 15;                        // N (tile) / A-row i
    const int kh    = lane >> 4;                        // K half within slice

    v8f acc = {};
#pragma unroll
    for (int kb = 0; kb < 8; ++kb) {
        const int j0 = 4 * kb + 2 * kh;                 // K index of element .x
        v2f a, bf;
        // A fragment: row i = col; band condition 0 <= j - i <= 15
        const int d0 = j0 - col;
        const int d1 = j0 + 1 - col;
        a.x = (d0 >= 0 && d0 < kW) ? 1.0f : 0.0f;
        a.y = (d1 >= 0 && d1 < kW) ? 1.0f : 0.0f;
        // B fragment: column n = col, rows j0, j0+1
        const int ridx = wbase + 16 * col + j0;
        bf.x = s_r[ridx];
        bf.y = s_r[ridx + 1];
        // (neg_a, A, neg_b, B, c_mod, C, reuse_a, reuse_b)
        acc = __builtin_amdgcn_wmma_f32_16x16x4_f32(
            false, a, false, bf, (short)0, acc, false, false);
    }

    // ---- Phase 3: scale by exp(logits[t]) and store ----
    // D element v -> output row i = kh*8 + v, tile n = col:
    //   t_local = wbase + 16*col + kh*8 + v   (8 consecutive floats per lane)
    const int tl = wbase + 16 * col + kh * 8;
#pragma unroll
    for (int v = 0; v < 8; ++v) {
        const float e = __expf(s_logits[tl + v + (kW - 1)]); // logits[c0+tl+v]
        row_out[c0 + tl + v] = e * acc[v];
    }
}

extern "C" void kernel_launch(void* const* d_in, const int* in_sizes, int n_in,
                              void* d_out, int out_size, void* d_ws, size_t ws_size,
                              hipStream_t stream) {
    (void)n_in; (void)out_size; (void)d_ws; (void)ws_size;
    const float* emit   = (const float*)d_in[0];   // emit_probs [B,T]
    const float* logits = (const float*)d_in[1];   // softmax_logits [B,T]
    float* out = (float*)d_out;

    const int B = in_sizes[0] / kT;                // 64 for the reference setup
    const int blocks = B * (kT / kSeg);            // 512
    mocha_chunk_attn_kernel<<<blocks, kThr, 0, stream>>>(emit, logits, out);
}